// Spectral_Filter_Transform_42588895707583
// MI455X (gfx1250) — compile-verified
//
#include <hip/hip_runtime.h>
#include <hip/hip_bf16.h>

// Fused spectral-topk-smooth for MI455X (gfx1250, wave32).
// Pipeline per (b,f) sequence of T=4096:
//   TDM (tensor_load_to_lds) DMA of the strided sequence into LDS,
//   4096-pt FFT as four-step 64x64 using V_WMMA_F32_16X16X4_F32,
//   top-16 magnitude bins (wave32 shfl argmax), direct 16-tone inverse
//   synthesis, flip-padded 25-tap Hamming smoothing. All staging in LDS.

typedef float v2f __attribute__((ext_vector_type(2)));
typedef float v8f __attribute__((ext_vector_type(8)));
typedef unsigned int uint32x4 __attribute__((ext_vector_type(4)));
typedef int int32x8 __attribute__((ext_vector_type(8)));
typedef int int32x4 __attribute__((ext_vector_type(4)));

#define T_LEN   4096
#define FDIM    128
#define WIN     25
#define HALF    12
#define KTOP    16
#define WAVES   4
#define TWO_PI  6.28318530717958647692f

// ---- WMMA helpers: 64-wide row-major A and B in LDS, f32 16x16x4 tiles ----
// A fragment (16x4): lanes 0-15 -> M=0..15 K={0,1}; lanes 16-31 -> K={2,3}.
// B fragment (4x16): lanes 0-15 -> N=0..15 K rows {0,1}; lanes 16-31 -> {2,3}.
// C/D (16x16): VGPR r -> row (r + 8*(lane>=16)), col = lane&15.

__device__ __forceinline__ void wmma_mm64(const float* __restrict__ A,
                                          const float* __restrict__ B,
                                          int i, int j, int lane, v8f& acc) {
  const int m  = lane & 15;
  const int kh = (lane >> 4) << 1;           // 0 or 2
  #pragma unroll
  for (int kk = 0; kk < 16; ++kk) {
    const int k0 = kk * 4 + kh;
    v2f a, b;
    a.x = A[(i * 16 + m) * 64 + k0];
    a.y = A[(i * 16 + m) * 64 + k0 + 1];
    b.x = B[(k0    ) * 64 + j * 16 + m];
    b.y = B[(k0 + 1) * 64 + j * 16 + m];
    acc = __builtin_amdgcn_wmma_f32_16x16x4_f32(
        /*neg_a=*/false, a, /*neg_b=*/false, b,
        /*c_mod=*/(short)0, acc, /*reuse_a=*/false, /*reuse_b=*/false);
  }
}

__device__ __forceinline__ void wmma_store(float* __restrict__ D, int dstride,
                                           int i, int j, int lane, const v8f& acc) {
  const int n    = lane & 15;
  const int roff = (lane >> 4) * 8;
  #pragma unroll
  for (int r = 0; r < 8; ++r)
    D[(i * 16 + r + roff) * dstride + j * 16 + n] = acc[r];
}

extern "C" __global__ void __launch_bounds__(WAVES * 32, 1)
spectral_topk_smooth_kernel(const float* __restrict__ x, float* __restrict__ out) {
  extern __shared__ float smem[];
  float* Gr      = smem;               // 4096: cos(2*pi*c*a/64)
  float* Gi      = Gr + 4096;          // 4096: -sin(2*pi*c*a/64)
  float* wtab    = Gi + 4096;          // 32: hamming taps + sum at [WIN]
  float* wavebuf = wtab + 32;          // WAVES * 3 * 4096

  const int tid  = threadIdx.x;
  const int lane = tid & 31;
  const int wave = tid >> 5;

  float* buf0 = wavebuf + wave * 3 * 4096; // M -> (Outr | Outi) -> conv out
  float* buf1 = buf0 + 4096;               // Yr/Zr -> mag -> xf
  float* buf2 = buf1 + 4096;               // Yi/Zi -> kept-bin table

  const int b  = blockIdx.x >> 5;          // 32 batches
  const int fg = blockIdx.x & 31;          // 32 groups of 4 features
  const int f  = fg * 4 + wave;

  const size_t base = (size_t)b * T_LEN * FDIM + (size_t)fg * 4;

  // ---- TDM: DMA this wave's sequence x[b, :, f] (stride 128 f32) into buf0.
  //      2D descriptor: tile 1 x 4096, tensor_dim0_stride = 128, 4-byte elems.
  //      Descriptor words forced wave-uniform via readfirstlane (D# is SGPRs). ----
  {
    const unsigned long long ga = (unsigned long long)(const void*)(x + base + wave);
    const unsigned int lds_off = __builtin_amdgcn_readfirstlane(
        (unsigned int)(unsigned long long)(__SIZE_TYPE__)buf0);
    const unsigned int ga_lo = __builtin_amdgcn_readfirstlane((unsigned int)ga);
    const unsigned int ga_hi = __builtin_amdgcn_readfirstlane((unsigned int)(ga >> 32));

    uint32x4 g0;
    g0[0] = 1u;                                   // count=1, is_restore=0, no gather
    g0[1] = lds_off;                              // lds_addr (bytes)
    g0[2] = ga_lo;                                // global_addr[31:0]
    g0[3] = (ga_hi & 0x01ffffffu) | 0x80000000u;  // global_addr[56:32] | type=2

    int32x8 g1;
    g1[0] = 0x00020000;            // data_size=2 (4 bytes); workgroup_mask=0
    g1[1] = 0x00010000;            // tensor_dim0 = 1 (bits[31:16])
    g1[2] = (int)(4096u << 16);    // tensor_dim1 = 4096 (low 16 in bits[31:16])
    g1[3] = 0x00010000;            // tile_dim0 = 1 (bits[31:16])
    g1[4] = 0x00001000;            // tile_dim1 = 4096 (bits[15:0]); tile_dim2=0
    g1[5] = FDIM;                  // tensor_dim0_stride = 128 (low 32)
    g1[6] = 0;
    g1[7] = 0;

    int32x4 gz4 = {};              // groups 2/3 unused (2D tensor)
    int32x8 gz8 = {};              // trailing group (clang-23 6-arg form)
    __builtin_amdgcn_tensor_load_to_lds(g0, g1, gz4, gz4, gz8, 0);
  }

  // ---- DFT64 matrices + Hamming window (block-cooperative, overlaps TDM) ----
  for (int idx = tid; idx < 4096; idx += WAVES * 32) {
    const int c = idx >> 6, a = idx & 63;
    float s, co;
    __sincosf(-TWO_PI * (float)((c * a) & 63) * (1.0f / 64.0f), &s, &co);
    Gr[idx] = co;
    Gi[idx] = s;                           // = -sin(2*pi*c*a/64)
  }
  if (tid < WIN) wtab[tid] = 0.54f - 0.46f * __cosf(TWO_PI * (float)tid / 25.0f);
  if (tid == 0) {
    float s = 0.f;
    for (int j = 0; j < WIN; ++j) s += 0.54f - 0.46f * __cosf(TWO_PI * (float)j / 25.0f);
    wtab[WIN] = s;
  }

  __builtin_amdgcn_s_wait_tensorcnt(0);    // this wave's M tile resident
  __syncthreads();                         // G/wtab visible block-wide

  // ---- Step 1: Y = G x M (complex = real matrix x 2) -> buf1, buf2 ----
  #pragma unroll 1
  for (int i = 0; i < 4; ++i)
    #pragma unroll 1
    for (int j = 0; j < 4; ++j) {
      v8f accR = {}, accI = {};
      wmma_mm64(Gr, buf0, i, j, lane, accR);
      wmma_mm64(Gi, buf0, i, j, lane, accI);
      wmma_store(buf1, 64, i, j, lane, accR);
      wmma_store(buf2, 64, i, j, lane, accI);
    }
  __syncthreads();

  // ---- Step 2: twiddle Z[c][q] = Y[c][q] * e^{-2*pi*i*c*q/4096} (in place) ----
  for (int idx = lane; idx < 4096; idx += 32) {
    const int c = idx >> 6, q = idx & 63;
    float s, co;
    __sincosf(-TWO_PI * (float)((c * q) & 4095) * (1.0f / 4096.0f), &s, &co);
    const float yr = buf1[idx], yi = buf2[idx];
    buf1[idx] = yr * co - yi * s;
    buf2[idx] = yr * s + yi * co;
  }
  __syncthreads();

  // ---- Step 3: Out[c][d] = sum_q Z[c][q]*G[q][d], only d<32 (bins 0..2047).
  //      Outr = Zr*Gr - Zi*Gi ; Outi = Zr*Gi + Zi*Gr (combine with VALU;
  //      f32 WMMA NEG only supports C-negate). Out -> buf0 halves (64x32). ----
  #pragma unroll 1
  for (int i = 0; i < 4; ++i)
    #pragma unroll 1
    for (int j = 0; j < 2; ++j) {
      v8f a0 = {}, a1 = {};
      wmma_mm64(buf1, Gr, i, j, lane, a0);   // Zr*Gr
      wmma_mm64(buf2, Gi, i, j, lane, a1);   // Zi*Gi
      v8f outr;
      #pragma unroll
      for (int r = 0; r < 8; ++r) outr[r] = a0[r] - a1[r];
      wmma_store(buf0, 32, i, j, lane, outr);

      v8f a2 = {}, a3 = {};
      wmma_mm64(buf1, Gi, i, j, lane, a2);   // Zr*Gi
      wmma_mm64(buf2, Gr, i, j, lane, a3);   // Zi*Gr
      v8f outi;
      #pragma unroll
      for (int r = 0; r < 8; ++r) outi[r] = a2[r] + a3[r];
      wmma_store(buf0 + 2048, 32, i, j, lane, outi);
    }
  __syncthreads();

  // ---- Magnitudes mag[k], k = c + 64*d -> buf1 (Z dead) ----
  for (int k = lane; k < 2048; k += 32) {
    const int c = k & 63, d = k >> 6;
    const float re = buf0[c * 32 + d];
    const float im = buf0[2048 + c * 32 + d];
    buf1[k] = re * re + im * im;
  }
  __syncthreads();

  // ---- Top-16 via 16x wave32 argmax; kept table -> buf2[0..47] ----
  for (int it = 0; it < KTOP; ++it) {
    float best = -1.0f;
    int   bi   = 0;
    for (int k = lane; k < 2048; k += 32) {
      const float m = buf1[k];
      if (m > best) { best = m; bi = k; }
    }
    #pragma unroll
    for (int off = 16; off > 0; off >>= 1) {
      const float ov = __shfl_xor(best, off, 32);
      const int   oi = __shfl_xor(bi, off, 32);
      if (ov > best || (ov == best && oi < bi)) { best = ov; bi = oi; }
    }
    if (lane == 0) {
      const int c = bi & 63, d = bi >> 6;
      buf2[it]          = buf0[c * 32 + d];          // re
      buf2[KTOP + it]   = buf0[2048 + c * 32 + d];   // im
      buf2[2*KTOP + it] = (float)bi;                 // bin index
      buf1[bi] = -1.0f;
    }
    __syncthreads();
  }

  // ---- Inverse synthesis: xf[t] = sum over kept bins (rotation recurrence).
  //      Each lane owns t in [lane*128, lane*128+128). xf -> buf1. ----
  {
    const int t0 = lane * 128;
    #pragma unroll 1
    for (int j = 0; j < 128; ++j) buf1[t0 + j] = 0.0f;
    #pragma unroll 1
    for (int kb = 0; kb < KTOP; ++kb) {
      const float cr  = buf2[kb];
      const float ci  = buf2[KTOP + kb];
      const int   k   = (int)buf2[2*KTOP + kb];
      const float wgt = (k == 0) ? 1.0f : 2.0f;
      float dS, dC, s, c;
      __sincosf(TWO_PI * (float)k * (1.0f / 4096.0f), &dS, &dC);
      __sincosf(TWO_PI * (float)((k * t0) & 4095) * (1.0f / 4096.0f), &s, &c);
      #pragma unroll 1
      for (int j = 0; j < 128; ++j) {
        buf1[t0 + j] += wgt * (cr * c - ci * s);
        const float nc = c * dC - s * dS;
        s = s * dC + c * dS;
        c = nc;
      }
    }
  }
  __syncthreads();

  // ---- Flip-padded 25-tap Hamming correlation -> buf0 (LDS staging) ----
  const float inv = 1.0f / (wtab[WIN] * 4096.0f);
  for (int t = lane; t < T_LEN; t += 32) {
    float acc = 0.0f;
    #pragma unroll
    for (int j = 0; j < WIN; ++j) {
      const int i   = t + j;
      const int src = (i < HALF) ? (HALF - 1 - i)
                    : (i < T_LEN + HALF) ? (i - HALF)
                    : (2 * T_LEN + 2 * HALF - 1 - i);   // 8215 - i
      acc += wtab[j] * buf1[src];
    }
    buf0[t] = acc * inv;
  }
  __syncthreads();

  // ---- Coalesced block-cooperative writeback: (t, f-local) pairs ----
  for (int it = 0; it < 128; ++it) {
    const int t  = (tid >> 2) + 32 * it;
    const int fl = tid & 3;
    out[base + (size_t)t * FDIM + fl] = (wavebuf + fl * 3 * 4096)[t];
  }
}

extern "C" void kernel_launch(void* const* d_in, const int* in_sizes, int n_in,
                              void* d_out, int out_size, void* d_ws, size_t ws_size,
                              hipStream_t stream) {
  (void)in_sizes; (void)n_in; (void)d_ws; (void)ws_size; (void)out_size;
  const float* x = (const float*)d_in[0];   // (32, 4096, 128) f32
  float* out = (float*)d_out;               // (32, 4096, 128) f32

  const size_t shmem = (size_t)(4096 * 2 + 32 + WAVES * 3 * 4096) * sizeof(float);
  (void)hipFuncSetAttribute((const void*)spectral_topk_smooth_kernel,
                            hipFuncAttributeMaxDynamicSharedMemorySize, (int)shmem);

  dim3 grid(32 * 32);        // batch * feature-groups-of-4
  dim3 block(WAVES * 32);    // 4 waves, one sequence each
  spectral_topk_smooth_kernel<<<grid, block, shmem, stream>>>(x, out);
}